// UnrolledLRU_37709812859308
// MI455X (gfx1250) — compile-verified
//
#include <hip/hip_runtime.h>
#include <hip/hip_bf16.h>

// Problem constants (match reference)
#define L_LEN 32768
#define N_DIM 256
#define H_DIM 256
#define T_CHUNK 256
#define C_CHUNKS (L_LEN / T_CHUNK)   // 128

typedef float v2f __attribute__((ext_vector_type(2)));
typedef float v8f __attribute__((ext_vector_type(8)));

// ---------------------------------------------------------------------------
// Workspace layout (floats):
//   lam_re[N], lam_im[N]
//   Wr[H*N]  : Wr[h*N+n] = B_re[n*H+h] * exp(gamma_log[n])   (B_norm transposed)
//   Wi[H*N]
//   Ctr[N*H] : Ctr[n*H+h] =  C_re[h*N+n]                     (C transposed)
//   Cti[N*H] : Cti[n*H+h] = -C_im[h*N+n]                     (pre-negated)
//   car_re[C*N], car_im[C*N]   (carry-in state for each chunk)
//   bu_re[L*N], bu_im[L*N]     (Bu, then scanned states, in place)
// ---------------------------------------------------------------------------

__global__ __launch_bounds__(256)
void lru_prep_kernel(const float* __restrict__ nu_log,
                     const float* __restrict__ theta_log,
                     const float* __restrict__ B_re,
                     const float* __restrict__ B_im,
                     const float* __restrict__ C_re,
                     const float* __restrict__ C_im,
                     const float* __restrict__ gamma_log,
                     float* __restrict__ lam_re, float* __restrict__ lam_im,
                     float* __restrict__ Wr, float* __restrict__ Wi,
                     float* __restrict__ Ctr, float* __restrict__ Cti) {
    int idx = blockIdx.x * blockDim.x + threadIdx.x;   // 0 .. N*H-1
    int n = idx / H_DIM;
    int h = idx % H_DIM;
    float g = __expf(gamma_log[n]);
    Wr[h * N_DIM + n] = B_re[n * H_DIM + h] * g;
    Wi[h * N_DIM + n] = B_im[n * H_DIM + h] * g;
    Ctr[n * H_DIM + h] =  C_re[h * N_DIM + n];
    Cti[n * H_DIM + h] = -C_im[h * N_DIM + n];
    if (h == 0) {
        float mag = __expf(-__expf(nu_log[n]));
        float th  = __expf(theta_log[n]);
        lam_re[n] = mag * __cosf(th);
        lam_im[n] = mag * __sinf(th);
    }
}

// ---------------------------------------------------------------------------
// GEMM1: Bu = x @ B_norm.T  (two real GEMMs, fp32 WMMA 16x16x4)
// Register-blocked: each wave computes a 64(M) x 16(N) strip = 4 M-tiles,
// so each B fragment (re & im) is loaded once and reused by 4 WMMAs.
// A fragment (16x4 f32): lane holds A[M = lane%16, K = 2*(lane/16)+v], v=0,1
// B fragment (4x16 f32): lane holds B[K = 2*(lane/16)+v, Ncol = lane%16]
// C/D (16x16 f32): vgpr r, lane holds D[M = r + 8*(lane/16), Ncol = lane%16]
// ---------------------------------------------------------------------------
__global__ __launch_bounds__(256)
void lru_gemm1_kernel(const float* __restrict__ x,
                      const float* __restrict__ Wr,
                      const float* __restrict__ Wi,
                      float* __restrict__ bu_re,
                      float* __restrict__ bu_im) {
    const int wave = threadIdx.x >> 5;
    const int lane = threadIdx.x & 31;
    const int idx  = blockIdx.x * 8 + wave;            // 8192 waves total
    const int tn   = idx & 15;                         // N tile (16 of them)
    const int mb   = idx >> 4;                         // 64-row M block (512)
    const int m0   = mb * 64;
    const int n0   = tn * 16;
    const int lg   = lane >> 4;                        // 0 or 1
    const int lm   = lane & 15;
    const int col  = n0 + lm;

    v8f accR[4] = {{}, {}, {}, {}};
    v8f accI[4] = {{}, {}, {}, {}};
    const float* xbase = x + (size_t)(m0 + lm) * H_DIM + 2 * lg;

    for (int k0 = 0; k0 < H_DIM; k0 += 4) {
        const int kb = (k0 + 2 * lg) * N_DIM + col;
        v2f br, bi;
        br.x = Wr[kb];
        br.y = Wr[kb + N_DIM];
        bi.x = Wi[kb];
        bi.y = Wi[kb + N_DIM];
#pragma unroll
        for (int mt = 0; mt < 4; ++mt) {
            v2f a = *(const v2f*)(xbase + (size_t)mt * 16 * H_DIM + k0);
            accR[mt] = __builtin_amdgcn_wmma_f32_16x16x4_f32(
                false, a, false, br, (short)0, accR[mt], false, false);
            accI[mt] = __builtin_amdgcn_wmma_f32_16x16x4_f32(
                false, a, false, bi, (short)0, accI[mt], false, false);
        }
    }

#pragma unroll
    for (int mt = 0; mt < 4; ++mt) {
        const int row0 = m0 + mt * 16 + 8 * lg;
#pragma unroll
        for (int r = 0; r < 8; ++r) {
            bu_re[(size_t)(row0 + r) * N_DIM + col] = accR[mt][r];
            bu_im[(size_t)(row0 + r) * N_DIM + col] = accI[mt][r];
        }
    }
}

// ---------------------------------------------------------------------------
// Scan phase 1: within-chunk local scan (in place). One thread per (chunk, n).
// ---------------------------------------------------------------------------
__global__ __launch_bounds__(256)
void lru_scan_local_kernel(const float* __restrict__ lam_re,
                           const float* __restrict__ lam_im,
                           float* __restrict__ bu_re,
                           float* __restrict__ bu_im) {
    const int c = blockIdx.x;            // chunk
    const int n = threadIdx.x;           // channel
    const float lr = lam_re[n], li = lam_im[n];
    float sr = 0.f, si = 0.f;
    size_t base = (size_t)c * T_CHUNK * N_DIM + n;
    for (int t = 0; t < T_CHUNK; ++t) {
        float br = bu_re[base], bi = bu_im[base];
        float nr = lr * sr - li * si + br;
        float ni = lr * si + li * sr + bi;
        sr = nr; si = ni;
        bu_re[base] = sr;
        bu_im[base] = si;
        base += N_DIM;
    }
}

// ---------------------------------------------------------------------------
// Scan phase 2: cross-chunk carry combine. One block of N threads;
// thread n walks the C=128 chunks sequentially. Lam^T via 8 squarings.
// ---------------------------------------------------------------------------
__global__ __launch_bounds__(256)
void lru_scan_carry_kernel(const float* __restrict__ lam_re,
                           const float* __restrict__ lam_im,
                           const float* __restrict__ bu_re,
                           const float* __restrict__ bu_im,
                           float* __restrict__ car_re,
                           float* __restrict__ car_im) {
    const int n = threadIdx.x;
    float pr = lam_re[n], pi = lam_im[n];
#pragma unroll
    for (int i = 0; i < 8; ++i) {        // Lam^(2^8) = Lam^256 = Lam^T
        float nr = pr * pr - pi * pi;
        pi = 2.f * pr * pi;
        pr = nr;
    }
    float cr = 0.f, ci = 0.f;
    for (int c = 0; c < C_CHUNKS; ++c) {
        car_re[c * N_DIM + n] = cr;
        car_im[c * N_DIM + n] = ci;
        size_t last = ((size_t)(c + 1) * T_CHUNK - 1) * N_DIM + n;
        float fr = bu_re[last], fi = bu_im[last];
        float nr = pr * cr - pi * ci + fr;
        float ni = pr * ci + pi * cr + fi;
        cr = nr; ci = ni;
    }
}

// ---------------------------------------------------------------------------
// Scan phase 3: apply carries: state_t += Lam^(tau+1) * carry_c  (tau = t in chunk)
// ---------------------------------------------------------------------------
__global__ __launch_bounds__(256)
void lru_scan_apply_kernel(const float* __restrict__ lam_re,
                           const float* __restrict__ lam_im,
                           const float* __restrict__ car_re,
                           const float* __restrict__ car_im,
                           float* __restrict__ bu_re,
                           float* __restrict__ bu_im) {
    const int c = blockIdx.x;
    const int n = threadIdx.x;
    const float lr = lam_re[n], li = lam_im[n];
    const float cr = car_re[c * N_DIM + n];
    const float ci = car_im[c * N_DIM + n];
    float pr = lr, pi = li;              // Lam^1 at tau = 0
    size_t base = (size_t)c * T_CHUNK * N_DIM + n;
    for (int t = 0; t < T_CHUNK; ++t) {
        bu_re[base] += pr * cr - pi * ci;
        bu_im[base] += pr * ci + pi * cr;
        float nr = pr * lr - pi * li;
        pi = pr * li + pi * lr;
        pr = nr;
        base += N_DIM;
    }
}

// ---------------------------------------------------------------------------
// GEMM2: y = states_re @ Ctr + states_im @ Cti (Cti pre-negated) + x * D
// Register-blocked 64x16 strip per wave; Ctr/Cti fragments reused 4x.
// ---------------------------------------------------------------------------
__global__ __launch_bounds__(256)
void lru_gemm2_kernel(const float* __restrict__ bu_re,
                      const float* __restrict__ bu_im,
                      const float* __restrict__ Ctr,
                      const float* __restrict__ Cti,
                      const float* __restrict__ x,
                      const float* __restrict__ Dv,
                      float* __restrict__ y) {
    const int wave = threadIdx.x >> 5;
    const int lane = threadIdx.x & 31;
    const int idx  = blockIdx.x * 8 + wave;            // 8192 waves total
    const int th   = idx & 15;                         // H tile
    const int mb   = idx >> 4;                         // 64-row M block
    const int m0   = mb * 64;
    const int h0   = th * 16;
    const int lg   = lane >> 4;
    const int lm   = lane & 15;
    const int col  = h0 + lm;

    v8f acc[4] = {{}, {}, {}, {}};
    const float* sbase = bu_re + (size_t)(m0 + lm) * N_DIM + 2 * lg;
    const float* ibase = bu_im + (size_t)(m0 + lm) * N_DIM + 2 * lg;

    for (int k0 = 0; k0 < N_DIM; k0 += 4) {
        const int kb = (k0 + 2 * lg) * H_DIM + col;
        v2f br, bi;
        br.x = Ctr[kb];
        br.y = Ctr[kb + H_DIM];
        bi.x = Cti[kb];
        bi.y = Cti[kb + H_DIM];
#pragma unroll
        for (int mt = 0; mt < 4; ++mt) {
            v2f ar = *(const v2f*)(sbase + (size_t)mt * 16 * N_DIM + k0);
            v2f ai = *(const v2f*)(ibase + (size_t)mt * 16 * N_DIM + k0);
            acc[mt] = __builtin_amdgcn_wmma_f32_16x16x4_f32(
                false, ar, false, br, (short)0, acc[mt], false, false);
            acc[mt] = __builtin_amdgcn_wmma_f32_16x16x4_f32(
                false, ai, false, bi, (short)0, acc[mt], false, false);
        }
    }

    const float d = Dv[col];
#pragma unroll
    for (int mt = 0; mt < 4; ++mt) {
        const int row0 = m0 + mt * 16 + 8 * lg;
#pragma unroll
        for (int r = 0; r < 8; ++r) {
            size_t o = (size_t)(row0 + r) * H_DIM + col;
            y[o] = acc[mt][r] + x[o] * d;
        }
    }
}

// ---------------------------------------------------------------------------
extern "C" void kernel_launch(void* const* d_in, const int* in_sizes, int n_in,
                              void* d_out, int out_size, void* d_ws, size_t ws_size,
                              hipStream_t stream) {
    const float* x         = (const float*)d_in[0];
    const float* nu_log    = (const float*)d_in[1];
    const float* theta_log = (const float*)d_in[2];
    const float* B_re      = (const float*)d_in[3];
    const float* B_im      = (const float*)d_in[4];
    const float* C_re      = (const float*)d_in[5];
    const float* C_im      = (const float*)d_in[6];
    const float* Dv        = (const float*)d_in[7];
    const float* gamma_log = (const float*)d_in[8];
    float* y = (float*)d_out;

    float* ws = (float*)d_ws;
    float* lam_re = ws;                       // N
    float* lam_im = lam_re + N_DIM;           // N
    float* Wr     = lam_im + N_DIM;           // H*N
    float* Wi     = Wr + H_DIM * N_DIM;       // H*N
    float* Ctr    = Wi + H_DIM * N_DIM;       // N*H
    float* Cti    = Ctr + N_DIM * H_DIM;      // N*H
    float* car_re = Cti + N_DIM * H_DIM;      // C*N
    float* car_im = car_re + C_CHUNKS * N_DIM;
    float* bu_re  = car_im + C_CHUNKS * N_DIM;              // L*N
    float* bu_im  = bu_re + (size_t)L_LEN * N_DIM;          // L*N

    // 1. Parameter prep / transposes
    lru_prep_kernel<<<(N_DIM * H_DIM) / 256, 256, 0, stream>>>(
        nu_log, theta_log, B_re, B_im, C_re, C_im, gamma_log,
        lam_re, lam_im, Wr, Wi, Ctr, Cti);

    // 2. Bu = x @ B_norm.T  (8192 waves, each a 64x16 strip)
    const int nwaves = (L_LEN / 64) * (N_DIM / 16);         // 8192
    lru_gemm1_kernel<<<nwaves / 8, 256, 0, stream>>>(x, Wr, Wi, bu_re, bu_im);

    // 3. Chunked associative scan
    lru_scan_local_kernel<<<C_CHUNKS, N_DIM, 0, stream>>>(lam_re, lam_im, bu_re, bu_im);
    lru_scan_carry_kernel<<<1, N_DIM, 0, stream>>>(lam_re, lam_im, bu_re, bu_im,
                                                   car_re, car_im);
    lru_scan_apply_kernel<<<C_CHUNKS, N_DIM, 0, stream>>>(lam_re, lam_im,
                                                          car_re, car_im, bu_re, bu_im);

    // 4. y = Re(states @ C.T) + x * D
    lru_gemm2_kernel<<<nwaves / 8, 256, 0, stream>>>(bu_re, bu_im, Ctr, Cti, x, Dv, y);
}